// HierarchicalVocabularyEmbedding_27066883900162
// MI455X (gfx1250) — compile-verified
//
#include <hip/hip_runtime.h>

#define EMBED 1024
#define TPB 128   // tokens per block

typedef float v2f __attribute__((ext_vector_type(2)));
typedef float v4f __attribute__((ext_vector_type(4)));
typedef float v8f __attribute__((ext_vector_type(8)));

__global__ __launch_bounds__(256) void adaptive_embed_wmma_f32(
    const int*   __restrict__ ids,
    const float* __restrict__ emb0,
    const float* __restrict__ emb1,
    const float* __restrict__ emb2,
    const float* __restrict__ emb3,
    const float* __restrict__ w1,
    const float* __restrict__ w2,
    const float* __restrict__ w3,
    float* __restrict__ out,
    int n_tok)
{
    __shared__ int   s_cnt[4];
    __shared__ short s_pos[4][TPB];   // token position within block; -1 = pad row
    __shared__ int   s_cid[4][TPB];   // row index within cluster's embedding table
    __shared__ float sA[16 * 260];    // staged A tile, row stride K+4 (bank-conflict-free)

    const int tid       = threadIdx.x;
    const int blockBase = blockIdx.x * TPB;

    if (tid < 4) s_cnt[tid] = 0;
    __syncthreads();

    // ---- Phase 1: classify tokens into compacted per-cluster lists ----
    if (tid < TPB) {
        int gt = blockBase + tid;
        if (gt < n_tok) {
            int id = ids[gt];
            int c, start;
            if (id < 20000)       { c = 0; start = 0; }
            else if (id < 100000) { c = 1; start = 20000; }
            else if (id < 299999) { c = 2; start = 100000; }
            else                  { c = 3; start = 299999; }
            int slot = atomicAdd(&s_cnt[c], 1);
            s_pos[c][slot] = (short)tid;
            s_cid[c][slot] = id - start;
        }
    }
    __syncthreads();

    // Pad cluster 1..3 lists up to a multiple of 16 rows
    for (int c = 1; c < 4; ++c) {
        int nc  = s_cnt[c];
        int ncp = (nc + 15) & ~15;
        for (int j = nc + tid; j < ncp; j += 256) {
            s_pos[c][j] = -1;
            s_cid[c][j] = 0;
        }
    }
    __syncthreads();

    // ---- Phase 2: cluster 0 = identity projection, straight row copy ----
    {
        int n0 = s_cnt[0];
        const v4f* e0 = reinterpret_cast<const v4f*>(emb0);
        v4f*       o4 = reinterpret_cast<v4f*>(out);
        for (int i = 0; i < n0; ++i) {
            int pos = s_pos[0][i];
            int cid = s_cid[0][i];
            v4f v = e0[(size_t)cid * (EMBED / 4) + tid];
            __builtin_nontemporal_store(v, &o4[(size_t)(blockBase + pos) * (EMBED / 4) + tid]);
        }
    }

    // ---- Phase 3: clusters 1..3 via V_WMMA_F32_16X16X4_F32 ----
    const int lane = tid & 31;
    const int wave = tid >> 5;     // 8 waves
    const int m    = lane & 15;    // row (A) / col (B,C)
    const int hi   = lane >> 4;    // lane half selects K pair / C row group

    for (int c = 1; c < 4; ++c) {
        const float* __restrict__ W;
        const float* __restrict__ E;
        int K;
        if (c == 1)      { W = w1; E = emb1; K = 256; }
        else if (c == 2) { W = w2; E = emb2; K = 64;  }
        else             { W = w3; E = emb3; K = 32;  }

        const int S      = K + 4;          // padded LDS row stride (floats)
        int       nc     = s_cnt[c];
        int       ntiles = (nc + 15) >> 4;

        for (int t = 0; t < ntiles; ++t) {
            const bool tail = (t == ntiles - 1) && ((nc & 15) != 0);

            __syncthreads();
            // Stage gathered A tile (16 rows x K f32) into LDS, 128-bit chunks
            {
                int r  = tid >> 4;        // row 0..15
                int lc = tid & 15;        // 16 threads per row
                int cid = s_cid[c][t * 16 + r];
                const v4f* src = reinterpret_cast<const v4f*>(E + (size_t)cid * K);
                v4f*       dst = reinterpret_cast<v4f*>(&sA[r * S]);
                for (int k4 = lc; k4 < (K >> 2); k4 += 16)
                    dst[k4] = src[k4];
            }
            __syncthreads();

            // Hoist this tile's 8 output-row positions (C layout: row = v + 8*hi)
            int pos8[8];
            #pragma unroll
            for (int v = 0; v < 8; ++v)
                pos8[v] = s_pos[c][t * 16 + v + (hi << 3)];

            // Each wave covers columns [wave*128, wave*128+128) = 8 col-tiles
            for (int ct = 0; ct < 8; ++ct) {
                int colbase = wave * 128 + ct * 16;
                // A frag source: lane L vgpr v holds A[m][k0 + 2*hi + v]
                const float* Al = &sA[m * S + 2 * hi];
                // B frag source: lane L vgpr v holds B[k0 + 2*hi + v][colbase + m]
                const float* Wl = W + (size_t)(2 * hi) * EMBED + colbase + m;

                v8f acc = {};
                for (int k0 = 0; k0 < K; k0 += 4) {
                    v2f a, b;
                    a.x = Al[k0];
                    a.y = Al[k0 + 1];
                    b.x = Wl[(size_t)k0 * EMBED];
                    b.y = Wl[(size_t)k0 * EMBED + EMBED];
                    acc = __builtin_amdgcn_wmma_f32_16x16x4_f32(
                        false, a, false, b, (short)0, acc, false, false);
                }

                // Scatter 16x16 C tile to the tokens' output rows
                if (!tail) {
                    #pragma unroll
                    for (int v = 0; v < 8; ++v) {
                        __builtin_nontemporal_store(
                            acc[v],
                            &out[(size_t)(blockBase + pos8[v]) * EMBED + colbase + m]);
                    }
                } else {
                    #pragma unroll
                    for (int v = 0; v < 8; ++v) {
                        if (pos8[v] >= 0) {
                            __builtin_nontemporal_store(
                                acc[v],
                                &out[(size_t)(blockBase + pos8[v]) * EMBED + colbase + m]);
                        }
                    }
                }
            }
        }
    }
}

extern "C" void kernel_launch(void* const* d_in, const int* in_sizes, int n_in,
                              void* d_out, int out_size, void* d_ws, size_t ws_size,
                              hipStream_t stream) {
    const int*   ids  = (const int*)  d_in[0];
    const float* emb0 = (const float*)d_in[1];
    const float* emb1 = (const float*)d_in[2];
    const float* emb2 = (const float*)d_in[3];
    const float* emb3 = (const float*)d_in[4];
    const float* w1   = (const float*)d_in[5];
    const float* w2   = (const float*)d_in[6];
    const float* w3   = (const float*)d_in[7];
    float* out = (float*)d_out;

    int n_tok  = in_sizes[0];                 // 8 * 4096 = 32768
    int blocks = (n_tok + TPB - 1) / TPB;     // 256

    adaptive_embed_wmma_f32<<<blocks, 256, 0, stream>>>(
        ids, emb0, emb1, emb2, emb3, w1, w2, w3, out, n_tok);
}